// OneDConvBlock_730144440414
// MI455X (gfx1250) — compile-verified
//
#include <hip/hip_runtime.h>

typedef __attribute__((ext_vector_type(2))) float v2f;
typedef __attribute__((ext_vector_type(8))) float v8f;

#define EPSV 1e-8f

constexpr int Bsz  = 4;    // batch
constexpr int BC   = 128;  // bottleneck channels
constexpr int HC   = 512;  // hidden channels
constexpr int DILC = 4;    // dilation

constexpr int TMM = 64;    // block tile M (2 waves x 2 accumulators x 16)
constexpr int TNN = 64;    // block tile N (4 waves x 16)
constexpr int AK1 = 132;   // gemm1 A LDS stride (K=128 + 4; mult of 4 for b128)
constexpr int BN  = 68;    // B LDS stride (N=64 + 4; mult of 4 for b128)
constexpr int KH2 = 64;    // gemm2 K chunk
constexpr int AK2 = 68;    // gemm2 A LDS stride (K=64 + 4)

// ---------- gfx1250 async global->LDS staging (guarded; sync fallback) ----------
#if defined(__has_builtin)
#  if __has_builtin(__builtin_amdgcn_global_load_async_to_lds_b128)
#    define HAS_ASYNC_LDS 1
#  endif
#endif
#ifndef HAS_ASYNC_LDS
#  define HAS_ASYNC_LDS 0
#endif

typedef int v4i_b __attribute__((vector_size(16)));
typedef __attribute__((address_space(1))) v4i_b* gv4_ptr;   // global
typedef __attribute__((address_space(3))) v4i_b* lv4_ptr;   // LDS

__device__ __forceinline__ void cp_b128(float* dstLds, const float* srcGlb) {
#if HAS_ASYNC_LDS
    __builtin_amdgcn_global_load_async_to_lds_b128((gv4_ptr)srcGlb, (lv4_ptr)dstLds,
                                                   0, 0);
#else
    *(float4*)dstLds = *(const float4*)srcGlb;
#endif
}

template <int N>
__device__ __forceinline__ void wait_async() {
#if HAS_ASYNC_LDS
#  if __has_builtin(__builtin_amdgcn_s_wait_asynccnt)
    __builtin_amdgcn_s_wait_asynccnt(N);
#  else
    asm volatile("s_wait_asynccnt %0" ::"i"(N) : "memory");
#  endif
#endif
}

__device__ __forceinline__ v8f wmma_f32(v2f a, v2f b, v8f c) {
    return __builtin_amdgcn_wmma_f32_16x16x4_f32(false, a, false, b, (short)0, c,
                                                 false, false);
}

// ---------------- GEMM1: h1 = prelu(W1 @ x + b1); per-column channel partial sums
// Requires L % 64 == 0 (true for the reference L=8000).
__global__ __launch_bounds__(256)
void k_gemm1(const float* __restrict__ x, const float* __restrict__ W1,
             const float* __restrict__ b1, const float* __restrict__ a1,
             float* __restrict__ h1, float* __restrict__ psum,
             float* __restrict__ psq, int L)
{
    __shared__ float Asub[TMM * AK1];   // [m][k] 64x128
    __shared__ float Bsub[128 * BN];    // [k][n] 128x64 row-major
    __shared__ float redS[4 * TNN];
    __shared__ float redQ[4 * TNN];

    const int tid   = threadIdx.x;
    const int lBase = blockIdx.x * TNN;
    const int hBase = blockIdx.y * TMM;
    const int b     = blockIdx.z;

    // stage A tile: W1[hBase+m][k] (rows of 128 contiguous floats)
    for (int e = tid; e < TMM * 128 / 4; e += 256) {
        int f = e * 4, m = f >> 7, k4 = f & 127;
        cp_b128(&Asub[m * AK1 + k4], &W1[(size_t)(hBase + m) * 128 + k4]);
    }
    // stage B tile row-major: Bsub[k][n] = x[b][k][lBase+n]
    for (int e = tid; e < 128 * TNN / 4; e += 256) {
        int f = e * 4, k = f >> 6, n4 = f & 63;
        cp_b128(&Bsub[k * BN + n4], &x[((size_t)(b * BC + k)) * L + lBase + n4]);
    }
    wait_async<0>();
    __syncthreads();

    const int lane = tid & 31;
    const int w    = tid >> 5;
    const int mw   = w & 1;        // 2 waves along M, each owns 32 rows (2 accs)
    const int nw   = w >> 1;       // 4 waves along N, 16 cols each
    const int half = lane >> 4;
    const int mr   = lane & 15;
    const int nCol = nw * 16 + mr;
    const int koff = half * 2;     // lanes 0-15: K+0,+1 ; lanes 16-31: K+2,+3

    v8f acc0 = {0.f,0.f,0.f,0.f,0.f,0.f,0.f,0.f};
    v8f acc1 = acc0;
    const int aRow0 = (mw * 32 + mr) * AK1;
    const int aRow1 = aRow0 + 16 * AK1;
    for (int kb = 0; kb < 128; kb += 4) {
        v2f af0 = *(const v2f*)&Asub[aRow0 + kb + koff];
        v2f af1 = *(const v2f*)&Asub[aRow1 + kb + koff];
        v2f bf  = { Bsub[(kb + koff) * BN + nCol],
                    Bsub[(kb + koff + 1) * BN + nCol] };
        acc0 = wmma_f32(af0, bf, acc0);
        acc1 = wmma_f32(af1, bf, acc1);
    }

    // epilogue: bias + prelu, store h1, per-column sums for cln1
    const float alpha = a1[0];
    float ps = 0.f, pq = 0.f;
    const int nG = lBase + nCol;
    for (int r = 0; r < 8; ++r) {
        int hG0 = hBase + mw * 32 + half * 8 + r;
        float v = acc0[r] + b1[hG0];
        v = (v >= 0.f) ? v : alpha * v;
        h1[((size_t)(b * HC + hG0)) * L + nG] = v;
        ps += v; pq += v * v;
        int hG1 = hG0 + 16;
        float u = acc1[r] + b1[hG1];
        u = (u >= 0.f) ? u : alpha * u;
        h1[((size_t)(b * HC + hG1)) * L + nG] = u;
        ps += u; pq += u * u;
    }
    redS[(mw * 2 + half) * TNN + nCol] = ps;
    redQ[(mw * 2 + half) * TNN + nCol] = pq;
    __syncthreads();
    if (tid < TNN) {
        float s = redS[tid] + redS[TNN + tid] + redS[2*TNN + tid] + redS[3*TNN + tid];
        float q = redQ[tid] + redQ[TNN + tid] + redQ[2*TNN + tid] + redQ[3*TNN + tid];
        psum[((size_t)(b * 8 + blockIdx.y)) * L + lBase + tid] = s;
        psq [((size_t)(b * 8 + blockIdx.y)) * L + lBase + tid] = q;
    }
}

// ---------------- cumulative scan: reduce G groups then prefix-scan over L
__global__ __launch_bounds__(256)
void k_scan(const float* __restrict__ psum, const float* __restrict__ psq,
            float* __restrict__ meanO, float* __restrict__ rstdO, int L, int G)
{
    __shared__ float ss[256];
    __shared__ float sq[256];
    const int tid = threadIdx.x;
    const int b   = blockIdx.x;
    float carryS = 0.f, carryQ = 0.f;
    for (int base = 0; base < L; base += 256) {
        int l = base + tid;
        float s = 0.f, q = 0.f;
        if (l < L) {
            for (int g = 0; g < G; ++g) {
                s += psum[((size_t)(b * G + g)) * L + l];
                q += psq [((size_t)(b * G + g)) * L + l];
            }
        }
        ss[tid] = s; sq[tid] = q;
        __syncthreads();
        for (int off = 1; off < 256; off <<= 1) {
            float as = 0.f, aq = 0.f;
            if (tid >= off) { as = ss[tid - off]; aq = sq[tid - off]; }
            __syncthreads();
            ss[tid] += as; sq[tid] += aq;
            __syncthreads();
        }
        if (l < L) {
            float cs = carryS + ss[tid];
            float cq = carryQ + sq[tid];
            float cnt  = (float)HC * (float)(l + 1);
            float mean = cs / cnt;
            float var  = cq / cnt - mean * mean;
            meanO[(size_t)b * L + l] = mean;
            rstdO[(size_t)b * L + l] = rsqrtf(var + EPSV);
        }
        carryS += ss[255];
        carryQ += sq[255];
        __syncthreads();
    }
}

// ---------------- normalize + dilated depthwise conv + prelu + partial sums
__global__ __launch_bounds__(256)
void k_dwconv(const float* __restrict__ h1, const float* __restrict__ mean1,
              const float* __restrict__ rstd1, const float* __restrict__ dw,
              const float* __restrict__ db, const float* __restrict__ a2,
              float* __restrict__ h2, float* __restrict__ psum,
              float* __restrict__ psq, int L)
{
    const int tid = threadIdx.x;
    const int l   = blockIdx.x * 256 + tid;
    const int g   = blockIdx.y;     // group of 32 h-channels
    const int b   = blockIdx.z;
    if (l >= L) return;

    const float alpha = a2[0];
    const bool hasM = (l >= DILC);
    const bool hasP = (l + DILC < L);
    const float m0v = mean1[(size_t)b * L + l];
    const float r0v = rstd1[(size_t)b * L + l];
    float mM = 0.f, rM = 0.f, mP = 0.f, rP = 0.f;
    if (hasM) { mM = mean1[(size_t)b * L + l - DILC]; rM = rstd1[(size_t)b * L + l - DILC]; }
    if (hasP) { mP = mean1[(size_t)b * L + l + DILC]; rP = rstd1[(size_t)b * L + l + DILC]; }

    float s = 0.f, q = 0.f;
    for (int hh = 0; hh < 32; ++hh) {
        int h = g * 32 + hh;
        const float* row = h1 + ((size_t)(b * HC + h)) * L;
        float y = dw[h * 3 + 1] * (row[l] - m0v) * r0v + db[h];
        if (hasM) y += dw[h * 3 + 0] * (row[l - DILC] - mM) * rM;
        if (hasP) y += dw[h * 3 + 2] * (row[l + DILC] - mP) * rP;
        y = (y >= 0.f) ? y : alpha * y;
        h2[((size_t)(b * HC + h)) * L + l] = y;
        s += y; q += y * y;
    }
    psum[((size_t)(b * 16 + g)) * L + l] = s;
    psq [((size_t)(b * 16 + g)) * L + l] = q;
}

// ---------------- row sums of Wout / Wskip (for cln fold into epilogue)
__global__ __launch_bounds__(256)
void k_rowsum(const float* __restrict__ Wout, const float* __restrict__ Wskip,
              float* __restrict__ rs)
{
    const int tid = threadIdx.x;
    const float* W = (tid < BC) ? Wout : Wskip;
    const int row  = tid & (BC - 1);
    float s = 0.f;
    for (int k = 0; k < HC; ++k) s += W[(size_t)row * HC + k];
    rs[tid] = s;
}

// ---------------- GEMM2: out = r*(Wout@h2) - r*m*rowsum + bout + x ; skip analog
// Double-buffered async K-chunks. Requires L % 64 == 0.
__global__ __launch_bounds__(256)
void k_gemm2(const float* __restrict__ h2, const float* __restrict__ mean2,
             const float* __restrict__ rstd2,
             const float* __restrict__ Wout, const float* __restrict__ bout,
             const float* __restrict__ Wskip, const float* __restrict__ bskip,
             const float* __restrict__ rowsums, const float* __restrict__ x,
             float* __restrict__ outBuf, int L)
{
    __shared__ float Asub[2][TMM * AK2];   // [m][k] 64x64 per buffer
    __shared__ float Bsub[2][KH2 * BN];    // [k][n] 64x64 per buffer

    const int tid    = threadIdx.x;
    const int lBase  = blockIdx.x * TNN;
    const int mBase  = blockIdx.y * TMM;
    const int zz     = blockIdx.z;
    const int b      = zz & 3;
    const bool isSkip = zz >= 4;
    const float* W    = isSkip ? Wskip : Wout;
    const float* bias = isSkip ? bskip : bout;
    const float* rs   = rowsums + (isSkip ? BC : 0);

    const int lane = tid & 31;
    const int w    = tid >> 5;
    const int mw   = w & 1;
    const int nw   = w >> 1;
    const int half = lane >> 4;
    const int mr   = lane & 15;
    const int nCol = nw * 16 + mr;
    const int koff = half * 2;
    const int aRow0 = (mw * 32 + mr) * AK2;
    const int aRow1 = aRow0 + 16 * AK2;

    // issue one K-chunk of staging (4 + 4 b128 async ops per thread)
    auto issue = [&](int chunk, int buf) {
        const int k0 = chunk * KH2;
        for (int e = tid; e < TMM * KH2 / 4; e += 256) {
            int f = e * 4, m = f >> 6, k4 = f & 63;
            cp_b128(&Asub[buf][m * AK2 + k4], &W[(size_t)(mBase + m) * HC + k0 + k4]);
        }
        for (int e = tid; e < KH2 * TNN / 4; e += 256) {
            int f = e * 4, k = f >> 6, n4 = f & 63;
            cp_b128(&Bsub[buf][k * BN + n4],
                    &h2[((size_t)(b * HC + k0 + k)) * L + lBase + n4]);
        }
    };

    v8f acc0 = {0.f,0.f,0.f,0.f,0.f,0.f,0.f,0.f};
    v8f acc1 = acc0;

    constexpr int NCH = HC / KH2;   // 8 chunks
    issue(0, 0);
    for (int c = 0; c < NCH; ++c) {
        const int buf = c & 1;
        if (c + 1 < NCH) {
            issue(c + 1, buf ^ 1);
            wait_async<8>();        // next chunk (8 ops/thread) may stay in flight
        } else {
            wait_async<0>();
        }
        __syncthreads();
        for (int kb = 0; kb < KH2; kb += 4) {
            v2f af0 = *(const v2f*)&Asub[buf][aRow0 + kb + koff];
            v2f af1 = *(const v2f*)&Asub[buf][aRow1 + kb + koff];
            v2f bf  = { Bsub[buf][(kb + koff) * BN + nCol],
                        Bsub[buf][(kb + koff + 1) * BN + nCol] };
            acc0 = wmma_f32(af0, bf, acc0);
            acc1 = wmma_f32(af1, bf, acc1);
        }
        __syncthreads();
    }

    const int nG   = lBase + nCol;
    const float m2v = mean2[(size_t)b * L + nG];
    const float r2v = rstd2[(size_t)b * L + nG];
    const size_t outOfs = isSkip ? (size_t)Bsz * BC * L : 0;
    for (int r = 0; r < 8; ++r) {
        int c0 = mBase + mw * 32 + half * 8 + r;
        float v = r2v * acc0[r] - r2v * m2v * rs[c0] + bias[c0];
        if (!isSkip) v += x[((size_t)(b * BC + c0)) * L + nG];
        outBuf[outOfs + ((size_t)(b * BC + c0)) * L + nG] = v;
        int c1 = c0 + 16;
        float u = r2v * acc1[r] - r2v * m2v * rs[c1] + bias[c1];
        if (!isSkip) u += x[((size_t)(b * BC + c1)) * L + nG];
        outBuf[outOfs + ((size_t)(b * BC + c1)) * L + nG] = u;
    }
}

extern "C" void kernel_launch(void* const* d_in, const int* in_sizes, int n_in,
                              void* d_out, int out_size, void* d_ws, size_t ws_size,
                              hipStream_t stream)
{
    const float* x     = (const float*)d_in[0];
    const float* W1    = (const float*)d_in[1];
    const float* b1    = (const float*)d_in[2];
    const float* dw    = (const float*)d_in[3];
    const float* db    = (const float*)d_in[4];
    const float* Wout  = (const float*)d_in[5];
    const float* bout  = (const float*)d_in[6];
    const float* Wskip = (const float*)d_in[7];
    const float* bskip = (const float*)d_in[8];
    const float* a1    = (const float*)d_in[9];
    const float* a2    = (const float*)d_in[10];

    const int L = in_sizes[0] / (Bsz * BC);

    float* ws = (float*)d_ws;
    size_t o = 0;
    float* h1    = ws + o; o += (size_t)Bsz * HC * L;
    float* h2    = ws + o; o += (size_t)Bsz * HC * L;
    float* psum1 = ws + o; o += (size_t)Bsz * 8 * L;
    float* psq1  = ws + o; o += (size_t)Bsz * 8 * L;
    float* mean1 = ws + o; o += (size_t)Bsz * L;
    float* rstd1 = ws + o; o += (size_t)Bsz * L;
    float* psum2 = ws + o; o += (size_t)Bsz * 16 * L;
    float* psq2  = ws + o; o += (size_t)Bsz * 16 * L;
    float* mean2 = ws + o; o += (size_t)Bsz * L;
    float* rstd2 = ws + o; o += (size_t)Bsz * L;
    float* rsums = ws + o; o += 2 * BC;

    k_rowsum<<<1, 256, 0, stream>>>(Wout, Wskip, rsums);

    dim3 g1(L / TNN, HC / TMM, Bsz);
    k_gemm1<<<g1, 256, 0, stream>>>(x, W1, b1, a1, h1, psum1, psq1, L);

    k_scan<<<Bsz, 256, 0, stream>>>(psum1, psq1, mean1, rstd1, L, 8);

    dim3 g3((L + 255) / 256, HC / 32, Bsz);
    k_dwconv<<<g3, 256, 0, stream>>>(h1, mean1, rstd1, dw, db, a2, h2, psum2, psq2, L);

    k_scan<<<Bsz, 256, 0, stream>>>(psum2, psq2, mean2, rstd2, L, 16);

    dim3 g5(L / TNN, BC / TMM, Bsz * 2);
    k_gemm2<<<g5, 256, 0, stream>>>(h2, mean2, rstd2, Wout, bout, Wskip, bskip,
                                    rsums, x, (float*)d_out, L);
}